// Jacobi2dUpdate_22479858827489
// MI455X (gfx1250) — compile-verified
//
#include <hip/hip_runtime.h>
#include <hip/hip_bf16.h>
#include <cstdint>

// Jacobi 2D step on (32,1,1024,1024) f32 grids for MI455X (gfx1250).
// Bandwidth-bound (AI ~ 0.5 flop/B): minimum HBM traffic = 12 B/elem
// => ~17 us at 23.3 TB/s. Optimize the data path:
//  - u halo tiles staged global->LDS with gfx1250 async DMA, B128-wide
//    (512 B per wave-instruction), tracked by ASYNCcnt
//  - streaming f/out use non-temporal hints to preserve L2 for u
//  - wave32-friendly 256-thread blocks, 128x16 tiles, conflict-free LDS

#define NX 1024
#define TX 128
#define TY 16
#define LDS_STRIDE 136                 // staged cols [c0-4, c0+132), 16B-aligned rows
#define LDS_ROWS 18                    // TY + 2 halo rows
#define CHUNKS_PER_ROW 34              // 136 floats / 4 per b128 chunk
#define HALO_CHUNKS (CHUNKS_PER_ROW * LDS_ROWS)   // 612
#define DUMMY_WORD (LDS_STRIDE * LDS_ROWS)        // 2448, 16B-aligned sink
#define LOAD_ITERS ((HALO_CHUNKS + 255) / 256)    // 3

__device__ __forceinline__ void async_load_b128_to_lds(unsigned lds_byte_off,
                                                       const float* base,
                                                       unsigned gl_byte_off) {
  // gfx1250: GLOBAL_LOAD_ASYNC_TO_LDS_B128 (GVS mode: saddr base + 32b voffset)
  asm volatile("global_load_async_to_lds_b128 %0, %1, %2"
               :
               : "v"(lds_byte_off), "v"(gl_byte_off), "s"(base)
               : "memory");
}

__device__ __forceinline__ void wait_asynccnt0() {
#if defined(__has_builtin) && __has_builtin(__builtin_amdgcn_s_wait_asynccnt)
  __builtin_amdgcn_s_wait_asynccnt(0);
#else
  asm volatile("s_wait_asynccnt 0" ::: "memory");
#endif
}

__global__ __launch_bounds__(256) void Jacobi2dUpdate_22479858827489_kernel(
    const float* __restrict__ f, const float* __restrict__ u,
    float* __restrict__ out) {
  __shared__ float tile[LDS_STRIDE * LDS_ROWS + 4];

  const int c0 = blockIdx.x * TX;
  const int r0 = blockIdx.y * TY;
  const size_t img_off = (size_t)blockIdx.z * NX * NX;
  const float* __restrict__ uimg = u + img_off;
  const float* __restrict__ fimg = f + img_off;
  float* __restrict__ oimg = out + img_off;

  const int tid = threadIdx.x;

  // ---- Stage u rows [r0-1, r0+16], cols [c0-4, c0+131] into LDS ----
  // c0 is a multiple of 128, so every 16B chunk aligns to a 4-float global
  // boundary: chunks never straddle row edges; invalid chunks go to a sink.
#pragma unroll
  for (int it = 0; it < LOAD_ITERS; ++it) {
    const int t = tid + it * 256;
    const int row = t / CHUNKS_PER_ROW;
    const int ch = t - row * CHUNKS_PER_ROW;
    const int gr = r0 - 1 + row;
    const int gc0 = c0 - 4 + ch * 4;        // first of 4 floats in this chunk
    const bool ok = (t < HALO_CHUNKS) & (gr >= 0) & (gr < NX) & (gc0 >= 0) &
                    (gc0 <= NX - 4);
    const int grc = min(max(gr, 0), NX - 1);
    const int gcc = min(max(gc0, 0), NX - 4);
    const int word = ok ? (row * LDS_STRIDE + ch * 4) : DUMMY_WORD;
    // Generic->flat cast truncated to 32 bits == wave-relative LDS offset.
    const unsigned lds_off = (unsigned)(uintptr_t)(&tile[word]);
    const unsigned gl_off = (unsigned)((grc * NX + gcc) * 4);
    async_load_b128_to_lds(lds_off, uimg, gl_off);
  }
  wait_asynccnt0();   // this wave's LDS writes landed
  __syncthreads();    // all waves' LDS writes visible

  // ---- Compute: fused stencil + Dirichlet/Neumann boundary remap ----
  const float cc = (1.0f / 1024.0f) * (1.0f / 1024.0f) * 0.25f;  // h*h/4

#pragma unroll
  for (int k = 0; k < (TX * TY) / 256; ++k) {
    const int e = tid + k * 256;
    const int lc = e & (TX - 1);
    const int lr = e >> 7;
    const int i = r0 + lr;
    const int j = c0 + lc;
    // Remapped source coords (all reads provably in-range):
    const int si = (i == 0) ? 1 : ((i == NX - 1) ? NX - 2 : i);
    const int sj = (j == 0) ? 1 : j;
    const int lR = si - (r0 - 1);
    const int lC = sj - c0 + 4;
    const float nN = tile[(lR - 1) * LDS_STRIDE + lC];
    const float nS = tile[(lR + 1) * LDS_STRIDE + lC];
    const float nW = tile[lR * LDS_STRIDE + (lC - 1)];
    const float nE = tile[lR * LDS_STRIDE + (lC + 1)];
    const float ff = __builtin_nontemporal_load(&fimg[si * NX + sj]);
    float v = 0.25f * (nN + nS + nW + nE) - cc * ff;
    v = (j == NX - 1) ? 0.0f : v;  // Dirichlet: zero last column
    __builtin_nontemporal_store(v, &oimg[i * NX + j]);
  }
}

extern "C" void kernel_launch(void* const* d_in, const int* in_sizes, int n_in,
                              void* d_out, int out_size, void* d_ws, size_t ws_size,
                              hipStream_t stream) {
  // setup_inputs order: f, dbc, nbc, u  (dbc/nbc unused by the reference math)
  const float* f = (const float*)d_in[0];
  const float* u = (const float*)d_in[3];
  float* out = (float*)d_out;
  const int batch = in_sizes[0] / (NX * NX);  // 32
  dim3 grid(NX / TX, NX / TY, batch);
  Jacobi2dUpdate_22479858827489_kernel<<<grid, 256, 0, stream>>>(f, u, out);
}